// MonetaBlock_28673201668257
// MI455X (gfx1250) — compile-verified
//
#include <hip/hip_runtime.h>
#include <hip/hip_bf16.h>
#include <math.h>

// ---------------------------------------------------------------------------
// MonetaBlock for MI455X (gfx1250, wave32).
//   B=4, SEQ=1024, D=256.
//   K1: qkv = x @ Wqkv^T          (WMMA f32 16x16x4, 16x64 strip per wave,
//                                  software-pipelined fragment loads)
//   K2: eta/alpha = sigmoid(x @ Wparam^T + b)
//   K3: causal dwconv(4) + RoPE + L2 norm  -> qh, kh, vh
//   K4: gpre = x @ Wgate^T        (WMMA)
//   K5: sequential scan, A (256x256 f32) resident in LDS, 1 WG per batch,
//       next-step vectors register-prefetched under the fused A-sweep
//   K6: out = (y @ Wout^T) * sigmoid(gpre)  (WMMA + epilogue)
// ---------------------------------------------------------------------------

#define BATCH 4
#define SEQ   1024
#define DIM   256
#define MTOT  (BATCH * SEQ)      // 4096 rows
#define EPSC  1e-6f

typedef __attribute__((ext_vector_type(2))) float v2f;
typedef __attribute__((ext_vector_type(8))) float v8f;

// ----------------------- WMMA GEMM: C = X @ W^T -----------------------------
// X: (M,DIM) row-major, W: (N,DIM) row-major, C: (M,N) row-major.
// One wave computes a 16x64 strip of C: one A-frag reused by 4 B-frags per
// k-step. Fragment loads are software-pipelined one k-step ahead with distinct
// registers so the four WMMAs of step k overlap the five loads of step k+4.
// A-frag (16x4 f32): lanes 0-15 -> M=lane, VGPR0/1=K0/K1; lanes 16-31 -> K2/K3.
// B-frag (4x16, B[k][n]=W[n][k]): lanes 0-15 -> N=lane, K0/K1; lanes 16-31 K2/K3.
// C/D (16x16 f32): VGPR r -> M=r (lanes 0-15, N=lane) / M=8+r (lanes 16-31).
__global__ void gemm_xwt_wmma(const float* __restrict__ X,
                              const float* __restrict__ W,
                              float* __restrict__ C,
                              int M, int N) {
  int wave = (int)((blockIdx.x * blockDim.x + threadIdx.x) >> 5);
  int lane = (int)(threadIdx.x & 31);
  int nstrips = N >> 6;                     // strips of 64 columns
  int tm = wave / nstrips;
  int ts = wave - tm * nstrips;
  if (tm * 16 >= M) return;                 // wave-uniform: EXEC stays all-1s
  int m0 = tm << 4, n0 = ts << 6;
  int half = lane >> 4;                     // 0: K0/K1, 1: K2/K3
  int l    = lane & 15;

  const float* xp = X + (size_t)(m0 + l) * DIM + 2 * half;
  const float* w0 = W + (size_t)(n0 +      l) * DIM + 2 * half;
  const float* w1 = W + (size_t)(n0 + 16 + l) * DIM + 2 * half;
  const float* w2 = W + (size_t)(n0 + 32 + l) * DIM + 2 * half;
  const float* w3 = W + (size_t)(n0 + 48 + l) * DIM + 2 * half;

  v8f acc0 = {}, acc1 = {}, acc2 = {}, acc3 = {};
  v2f a  = *(const v2f*)(xp);
  v2f b0 = *(const v2f*)(w0);
  v2f b1 = *(const v2f*)(w1);
  v2f b2 = *(const v2f*)(w2);
  v2f b3 = *(const v2f*)(w3);
#pragma unroll 4
  for (int k = 0; k < DIM; k += 4) {
    int kn = (k + 4 < DIM) ? k + 4 : 0;     // wrap: keeps look-ahead in-bounds
    v2f an  = *(const v2f*)(xp + kn);
    v2f bn0 = *(const v2f*)(w0 + kn);
    v2f bn1 = *(const v2f*)(w1 + kn);
    v2f bn2 = *(const v2f*)(w2 + kn);
    v2f bn3 = *(const v2f*)(w3 + kn);
    acc0 = __builtin_amdgcn_wmma_f32_16x16x4_f32(false, a, false, b0, (short)0, acc0, false, false);
    acc1 = __builtin_amdgcn_wmma_f32_16x16x4_f32(false, a, false, b1, (short)0, acc1, false, false);
    acc2 = __builtin_amdgcn_wmma_f32_16x16x4_f32(false, a, false, b2, (short)0, acc2, false, false);
    acc3 = __builtin_amdgcn_wmma_f32_16x16x4_f32(false, a, false, b3, (short)0, acc3, false, false);
    a = an; b0 = bn0; b1 = bn1; b2 = bn2; b3 = bn3;
  }
  float* crow = C + (size_t)(m0 + half * 8) * N + n0 + l;
#pragma unroll
  for (int r = 0; r < 8; ++r) {
    crow[(size_t)r * N +  0] = acc0[r];
    crow[(size_t)r * N + 16] = acc1[r];
    crow[(size_t)r * N + 32] = acc2[r];
    crow[(size_t)r * N + 48] = acc3[r];
  }
}

// --------------------- WMMA GEMM + gating epilogue --------------------------
// Out[m][n] = (Y @ Wout^T)[m][n] * sigmoid(Gpre[m][n])
__global__ void gemm_out_gate_wmma(const float* __restrict__ Y,
                                   const float* __restrict__ Wout,
                                   const float* __restrict__ Gpre,
                                   float* __restrict__ Out,
                                   int M, int N) {
  int wave = (int)((blockIdx.x * blockDim.x + threadIdx.x) >> 5);
  int lane = (int)(threadIdx.x & 31);
  int nstrips = N >> 6;
  int tm = wave / nstrips;
  int ts = wave - tm * nstrips;
  if (tm * 16 >= M) return;
  int m0 = tm << 4, n0 = ts << 6;
  int half = lane >> 4;
  int l    = lane & 15;

  const float* yp = Y    + (size_t)(m0 + l) * DIM + 2 * half;
  const float* w0 = Wout + (size_t)(n0 +      l) * DIM + 2 * half;
  const float* w1 = Wout + (size_t)(n0 + 16 + l) * DIM + 2 * half;
  const float* w2 = Wout + (size_t)(n0 + 32 + l) * DIM + 2 * half;
  const float* w3 = Wout + (size_t)(n0 + 48 + l) * DIM + 2 * half;

  v8f acc0 = {}, acc1 = {}, acc2 = {}, acc3 = {};
  v2f a  = *(const v2f*)(yp);
  v2f b0 = *(const v2f*)(w0);
  v2f b1 = *(const v2f*)(w1);
  v2f b2 = *(const v2f*)(w2);
  v2f b3 = *(const v2f*)(w3);
#pragma unroll 4
  for (int k = 0; k < DIM; k += 4) {
    int kn = (k + 4 < DIM) ? k + 4 : 0;
    v2f an  = *(const v2f*)(yp + kn);
    v2f bn0 = *(const v2f*)(w0 + kn);
    v2f bn1 = *(const v2f*)(w1 + kn);
    v2f bn2 = *(const v2f*)(w2 + kn);
    v2f bn3 = *(const v2f*)(w3 + kn);
    acc0 = __builtin_amdgcn_wmma_f32_16x16x4_f32(false, a, false, b0, (short)0, acc0, false, false);
    acc1 = __builtin_amdgcn_wmma_f32_16x16x4_f32(false, a, false, b1, (short)0, acc1, false, false);
    acc2 = __builtin_amdgcn_wmma_f32_16x16x4_f32(false, a, false, b2, (short)0, acc2, false, false);
    acc3 = __builtin_amdgcn_wmma_f32_16x16x4_f32(false, a, false, b3, (short)0, acc3, false, false);
    a = an; b0 = bn0; b1 = bn1; b2 = bn2; b3 = bn3;
  }
  size_t base = (size_t)(m0 + half * 8) * N + n0 + l;
#pragma unroll
  for (int r = 0; r < 8; ++r) {
    size_t o = base + (size_t)r * N;
    float g0 = Gpre[o +  0], g1 = Gpre[o + 16], g2 = Gpre[o + 32], g3 = Gpre[o + 48];
    Out[o +  0] = acc0[r] * (1.f / (1.f + __expf(-g0)));
    Out[o + 16] = acc1[r] * (1.f / (1.f + __expf(-g1)));
    Out[o + 32] = acc2[r] * (1.f / (1.f + __expf(-g2)));
    Out[o + 48] = acc3[r] * (1.f / (1.f + __expf(-g3)));
  }
}

// --------------------------- eta / alpha ------------------------------------
__global__ void params_kernel(const float* __restrict__ X,
                              const float* __restrict__ Wparam,
                              const float* __restrict__ bparam,
                              float* __restrict__ eta,
                              float* __restrict__ alpha, int M) {
  int m = (int)(blockIdx.x * blockDim.x + threadIdx.x);
  if (m >= M) return;
  const float* xr = X + (size_t)m * DIM;
  float s0 = 0.f, s1 = 0.f;
  for (int k = 0; k < DIM; ++k) {
    float xv = xr[k];
    s0 = fmaf(xv, Wparam[k], s0);
    s1 = fmaf(xv, Wparam[DIM + k], s1);
  }
  s0 += bparam[0];
  s1 += bparam[1];
  eta[m]   = 1.f / (1.f + __expf(-s0));
  alpha[m] = 1.f / (1.f + __expf(-s1));
}

// ----------------- causal dwconv(4) + RoPE + L2 norm ------------------------
// One block per (b,t) token. 128 threads; thread c handles channel pair (2c,2c+1).
__global__ void conv_rope_norm(const float* __restrict__ qkv,
                               const float* __restrict__ qw, const float* __restrict__ qb,
                               const float* __restrict__ kw, const float* __restrict__ kb,
                               const float* __restrict__ vw, const float* __restrict__ vb,
                               const float* __restrict__ fcos, const float* __restrict__ fsin,
                               float* __restrict__ qh, float* __restrict__ kh,
                               float* __restrict__ vh) {
  int bt = (int)blockIdx.x;           // b*SEQ + t
  int t  = bt & (SEQ - 1);
  int c  = (int)threadIdx.x;          // 0..127 rope-pair index
  int ch0 = 2 * c, ch1 = ch0 + 1;

  float q0 = qb[ch0], q1 = qb[ch1];
  float k0 = kb[ch0], k1 = kb[ch1];
  float v0 = vb[ch0], v1 = vb[ch1];
#pragma unroll
  for (int j = 0; j < 4; ++j) {
    int tt = t + j - 3;
    if (tt < 0) continue;
    const float* row = qkv + (size_t)(bt + j - 3) * (3 * DIM);
    q0 = fmaf(row[ch0],           qw[ch0 * 4 + j], q0);
    q1 = fmaf(row[ch1],           qw[ch1 * 4 + j], q1);
    k0 = fmaf(row[DIM + ch0],     kw[ch0 * 4 + j], k0);
    k1 = fmaf(row[DIM + ch1],     kw[ch1 * 4 + j], k1);
    v0 = fmaf(row[2 * DIM + ch0], vw[ch0 * 4 + j], v0);
    v1 = fmaf(row[2 * DIM + ch1], vw[ch1 * 4 + j], v1);
  }
  float cs = fcos[(size_t)t * (DIM / 2) + c];
  float sn = fsin[(size_t)t * (DIM / 2) + c];
  float qr = q0 * cs - q1 * sn, qi = q0 * sn + q1 * cs;
  float kr = k0 * cs - k1 * sn, ki = k0 * sn + k1 * cs;

  __shared__ float redq[128], redk[128];
  redq[c] = qr * qr + qi * qi;
  redk[c] = kr * kr + ki * ki;
  __syncthreads();
  for (int s = 64; s > 0; s >>= 1) {
    if (c < s) { redq[c] += redq[c + s]; redk[c] += redk[c + s]; }
    __syncthreads();
  }
  float qin = 1.f / fmaxf(sqrtf(redq[0]), 1e-12f);
  float kin = 1.f / fmaxf(sqrtf(redk[0]), 1e-12f);

  size_t o = (size_t)bt * DIM;
  qh[o + ch0] = qr * qin;  qh[o + ch1] = qi * qin;
  kh[o + ch0] = kr * kin;  kh[o + ch1] = ki * kin;
  vh[o + ch0] = v0;        vh[o + ch1] = v1;
}

// ------------------------------- the scan -----------------------------------
// One workgroup per batch; A (256x256 f32 = 256KB) lives in LDS for all 1024
// steps. Per step a single fused sweep of A does: update, Sum(A^4), y = q.A_new,
// pred_next = k_{t+1}.A_new (division by denom applied afterwards).
// Next-step q/k/v/eta/alpha are register-prefetched so their global-load
// latency is hidden under the ~1800-VALU-op sweep instead of sitting on the
// serial critical path.
__global__ __launch_bounds__(256) void scan_kernel(
    const float* __restrict__ qh, const float* __restrict__ kh,
    const float* __restrict__ vh, const float* __restrict__ eta,
    const float* __restrict__ alpha, const float* __restrict__ W0,
    float* __restrict__ ybuf) {
  extern __shared__ float smem[];
  float* A     = smem;              // 65536 floats
  float* sQ    = smem + 65536;      // 256
  float* sKc   = sQ    + DIM;       // 256 (k_t)
  float* sKn   = sKc   + DIM;       // 256 (k_{t+1})
  float* sV    = sKn   + DIM;       // 256
  float* sG    = sV    + DIM;       // 256 (grad_pred)
  float* sPred = sG    + DIM;       // 256 (raw pred for current step)
  float* pY    = sPred + DIM;       // 4*256 partials
  float* pP    = pY    + 4 * DIM;   // 4*256 partials
  float* sRed  = pP    + 4 * DIM;   // [0]=Sum A^4, [1]=1/denom

  int b   = (int)blockIdx.x;
  int tid = (int)threadIdx.x;
  int c   = tid & 63;               // column group: cols 4c..4c+3
  int r   = tid >> 6;               // row group: rows r*64..r*64+63
  const size_t bbase = (size_t)b * SEQ * DIM;
  const size_t ebase = (size_t)b * SEQ;

  float4* A4 = (float4*)A;
  for (int j = 0; j < 64; ++j)
    A4[(size_t)(r * 64 + j) * 64 + c] = make_float4(0.f, 0.f, 0.f, 0.f);
  if (tid < 2) sRed[tid] = 0.f;
  sKn[tid] = kh[bbase + tid];       // k_0, staged for the W0 matvec
  __syncthreads();

  {   // pred_0 = k_0 @ W0  (W0 = identity in the reference, applied honestly)
    float acc = 0.f;
    for (int d = 0; d < DIM; ++d) acc = fmaf(sKn[d], W0[(size_t)d * DIM + tid], acc);
    sPred[tid] = acc;
  }

  // prefetch registers for step t=0
  float rq  = qh[bbase + tid];
  float rk  = kh[bbase + tid];
  float rv  = vh[bbase + tid];
  float rkn = kh[bbase + DIM + tid];          // k_1 (SEQ > 1)
  float ret = eta[ebase];
  float ral = alpha[ebase];
  __syncthreads();

  for (int t = 0; t < SEQ; ++t) {
    size_t row = bbase + (size_t)t * DIM;
    // deposit prefetched step-t vectors into LDS
    sQ[tid]  = rq;
    sKc[tid] = rk;
    sV[tid]  = rv;
    sKn[tid] = rkn;
    float et = ret;
    float al = ral;
    // issue prefetch for step t+1 (consumed a full A-sweep later)
    if (t + 1 < SEQ) {
      rq  = qh[row + DIM + tid];
      rk  = kh[row + DIM + tid];
      rv  = vh[row + DIM + tid];
      rkn = (t + 2 < SEQ) ? kh[row + 2 * DIM + tid] : 0.f;
      ret = eta[ebase + t + 1];
      ral = alpha[ebase + t + 1];
    }
    __syncthreads();

    {   // grad_pred = 3 * tanh(10*diff) * diff^2
      float diff = sPred[tid] - sV[tid];
      sG[tid] = 3.f * tanhf(10.f * diff) * diff * diff;
    }
    __syncthreads();

    // fused sweep of A
    float4 g4 = ((const float4*)sG)[c];
    float4 ay = make_float4(0.f, 0.f, 0.f, 0.f);
    float4 ap = make_float4(0.f, 0.f, 0.f, 0.f);
    float  a4s = 0.f;
    for (int j = 0; j < 64; ++j) {
      int d = r * 64 + j;
      float ek = sKc[d] * et;
      float qd = sQ[d];
      float nd = sKn[d];
      float4 a = A4[(size_t)d * 64 + c];
      a.x = fmaf(al, a.x, -ek * g4.x);
      a.y = fmaf(al, a.y, -ek * g4.y);
      a.z = fmaf(al, a.z, -ek * g4.z);
      a.w = fmaf(al, a.w, -ek * g4.w);
      A4[(size_t)d * 64 + c] = a;
      float x0 = a.x * a.x, x1 = a.y * a.y, x2 = a.z * a.z, x3 = a.w * a.w;
      a4s = fmaf(x0, x0, a4s); a4s = fmaf(x1, x1, a4s);
      a4s = fmaf(x2, x2, a4s); a4s = fmaf(x3, x3, a4s);
      ay.x = fmaf(qd, a.x, ay.x); ay.y = fmaf(qd, a.y, ay.y);
      ay.z = fmaf(qd, a.z, ay.z); ay.w = fmaf(qd, a.w, ay.w);
      ap.x = fmaf(nd, a.x, ap.x); ap.y = fmaf(nd, a.y, ap.y);
      ap.z = fmaf(nd, a.z, ap.z); ap.w = fmaf(nd, a.w, ap.w);
    }
    ((float4*)pY)[r * 64 + c] = ay;
    ((float4*)pP)[r * 64 + c] = ap;

    // wave reduce Sum(A^4) then one LDS float atomic per wave
    for (int off = 16; off > 0; off >>= 1) a4s += __shfl_down(a4s, off, 32);
    if ((tid & 31) == 0) atomicAdd(&sRed[0], a4s);
    __syncthreads();

    if (tid == 0) {
      float s = sRed[0];
      sRed[1] = 1.f / (sqrtf(s) + EPSC);
      sRed[0] = 0.f;
    }
    __syncthreads();

    float invd = sRed[1];
    float yv = pY[tid] + pY[DIM + tid] + pY[2 * DIM + tid] + pY[3 * DIM + tid];
    float pv = pP[tid] + pP[DIM + tid] + pP[2 * DIM + tid] + pP[3 * DIM + tid];
    ybuf[row + tid] = yv * invd;       // y_t = q_t . A_new / denom
    sPred[tid] = pv * invd;            // pred_{t+1} = k_{t+1} . A_new / denom
    __syncthreads();
  }
}

// ---------------------------------------------------------------------------
extern "C" void kernel_launch(void* const* d_in, const int* in_sizes, int n_in,
                              void* d_out, int out_size, void* d_ws, size_t ws_size,
                              hipStream_t stream) {
  const float* x      = (const float*)d_in[0];
  const float* fcos   = (const float*)d_in[1];
  const float* fsin   = (const float*)d_in[2];
  const float* Wqkv   = (const float*)d_in[3];
  const float* qw     = (const float*)d_in[4];
  const float* qb     = (const float*)d_in[5];
  const float* kw     = (const float*)d_in[6];
  const float* kb     = (const float*)d_in[7];
  const float* vw     = (const float*)d_in[8];
  const float* vb     = (const float*)d_in[9];
  const float* Wparam = (const float*)d_in[10];
  const float* bparam = (const float*)d_in[11];
  const float* W0     = (const float*)d_in[12];
  const float* Wgate  = (const float*)d_in[13];
  const float* Wout   = (const float*)d_in[14];
  float* out = (float*)d_out;

  // workspace layout (floats): ~33.6 MB total
  float* ws   = (float*)d_ws;
  float* qkv  = ws;                                // 4096*768
  float* qh   = qkv  + (size_t)MTOT * 3 * DIM;     // 4096*256
  float* kh   = qh   + (size_t)MTOT * DIM;
  float* vh   = kh   + (size_t)MTOT * DIM;
  float* etab = vh   + (size_t)MTOT * DIM;         // 4096
  float* alpb = etab + MTOT;                       // 4096
  float* gpre = alpb + MTOT;                       // 4096*256
  float* ybuf = gpre + (size_t)MTOT * DIM;         // 4096*256

  // K1: qkv = x @ Wqkv^T   (M=4096, N=768, K=256) -> 3072 strip-waves
  {
    int waves = (MTOT / 16) * ((3 * DIM) / 64);
    gemm_xwt_wmma<<<waves / 4, 128, 0, stream>>>(x, Wqkv, qkv, MTOT, 3 * DIM);
  }
  // K2: eta/alpha
  params_kernel<<<MTOT / 256, 256, 0, stream>>>(x, Wparam, bparam, etab, alpb, MTOT);
  // K3: conv + rope + l2norm
  conv_rope_norm<<<MTOT, 128, 0, stream>>>(qkv, qw, qb, kw, kb, vw, vb,
                                           fcos, fsin, qh, kh, vh);
  // K4: gpre = x @ Wgate^T  (M=4096, N=256, K=256)
  {
    int waves = (MTOT / 16) * (DIM / 64);
    gemm_xwt_wmma<<<waves / 4, 128, 0, stream>>>(x, Wgate, gpre, MTOT, DIM);
  }
  // K5: scan — A resident in LDS (276,488 bytes dynamic LDS < 320KB WGP limit)
  {
    size_t smem = (size_t)(65536 + 6 * DIM + 8 * DIM + 2) * sizeof(float);
    (void)hipFuncSetAttribute((const void*)scan_kernel,
                              hipFuncAttributeMaxDynamicSharedMemorySize,
                              (int)smem);
    scan_kernel<<<BATCH, 256, smem, stream>>>(qh, kh, vh, etab, alpb, W0, ybuf);
  }
  // K6: out = (y @ Wout^T) * sigmoid(gpre)
  {
    int waves = (MTOT / 16) * (DIM / 64);
    gemm_out_gate_wmma<<<waves / 4, 128, 0, stream>>>(ybuf, Wout, gpre, out,
                                                      MTOT, DIM);
  }
}